// CopyMechanism_24713241821748
// MI455X (gfx1250) — compile-verified
//
#include <hip/hip_runtime.h>
#include <hip/hip_bf16.h>

#define HID 1024
#define BATCH 8
#define GDIM 512
#define CDIM 4096

typedef float v2f __attribute__((ext_vector_type(2)));
typedef float v8f __attribute__((ext_vector_type(8)));

// ---------------------------------------------------------------------------
// Kernel 1: dual GEMV via V_WMMA_F32_16X16X4_F32.
// s_out[r] = dot(X[r,:], wa), g_out[r] = dot(X[r,:], wg) for M rows (M%128==0).
//  - LDS holds wa, wg, and a zero row; each lane resolves its B-operand base
//    pointer ONCE (column 0 -> wa, column 1 -> wg, columns 2-15 -> zeros), so
//    the inner loop is branch-free: ds_load_b64 (A), ds_load_b64 (B), v_wmma.
//  - each wave owns a 16-row tile; stages 16x32 fp32 chunks with coalesced
//    128B row loads (stride-34 padding -> conflict-free transposed reads)
// ---------------------------------------------------------------------------
__global__ void gemv2_wmma_kernel(const float* __restrict__ X,
                                  const float* __restrict__ wa,
                                  const float* __restrict__ wg,
                                  float* __restrict__ s_out,
                                  float* __restrict__ g_out) {
    __shared__ float wsh[3][HID];        // [0]=wa  [1]=wg  [2]=zeros
    __shared__ float tile[8][16 * 34];   // 8 waves/block, stride-34 padded

    const int tid = threadIdx.x;
    for (int i = tid; i < HID; i += 256) {
        wsh[0][i] = wa[i];
        wsh[1][i] = wg[i];
        wsh[2][i] = 0.f;
    }
    __syncthreads();

    const int lane = tid & 31;
    const int wave = tid >> 5;
    const int row0 = (blockIdx.x * 8 + wave) * 16;

    const int m    = lane & 15;          // A-matrix row / B-matrix column
    const int half = lane >> 4;          // selects K pair {0,1} vs {2,3}

    // Per-lane B base pointer, resolved once (no branches in the hot loop).
    const int wsel = (m == 0) ? 0 : ((m == 1) ? 1 : 2);
    const float* bbase = wsh[wsel];

    const float* tw = tile[wave];
    float* twst = tile[wave];
    const int aoff = m * 34;             // loop-invariant part of A address

    v8f acc = {0.f, 0.f, 0.f, 0.f, 0.f, 0.f, 0.f, 0.f};

    for (int k0 = 0; k0 < HID; k0 += 32) {
        // Stage 16 rows x 32 cols; each row is one coalesced 128B load.
        #pragma unroll
        for (int r = 0; r < 16; ++r) {
            twst[r * 34 + lane] = X[(size_t)(row0 + r) * HID + (k0 + lane)];
        }
        // Wave-local LDS producer/consumer: DS ops in-order per wave.
        #pragma unroll
        for (int kk = 0; kk < 8; ++kk) {
            const int kloc = kk * 4 + 2 * half;          // even -> 8B aligned
            // A 16x4 fp32: lane m holds A[m][kloc], A[m][kloc+1]
            v2f a = *(const v2f*)(tw + aoff + kloc);
            // B 4x16 fp32: one unconditional ds_load_b64 per lane
            v2f b = *(const v2f*)(bbase + k0 + kloc);
            acc = __builtin_amdgcn_wmma_f32_16x16x4_f32(
                /*neg_a=*/false, a, /*neg_b=*/false, b,
                /*c_mod=*/(short)0, acc, /*reuse_a=*/false, /*reuse_b=*/false);
        }
    }

    // C/D layout: VGPR j, lanes 0-15 -> M=j N=lane; lanes 16-31 -> M=j+8.
    // Column 0 (attn) lives in lanes 0,16; column 1 (gate) in lanes 1,17.
    if (m < 2) {
        #pragma unroll
        for (int j = 0; j < 8; ++j) {
            const int row = row0 + j + half * 8;
            const float v = acc[j];
            if (m == 0) s_out[row] = v;
            else        g_out[row] = v;
        }
    }
}

// ---------------------------------------------------------------------------
// Kernel 2: per-batch masked softmax over C=4096, in place over sc.
// All-masked batch -> sum==0 -> 0 * inf = NaN everywhere (matches jax).
// ---------------------------------------------------------------------------
__global__ void softmax_mask_kernel(float* __restrict__ sc,
                                    const int* __restrict__ mask) {
    __shared__ float red[256];
    const int b   = blockIdx.x;
    const int tid = threadIdx.x;
    const float NEG_INF = -__builtin_inff();

    float mx = NEG_INF;
    for (int c = tid; c < CDIM; c += 256) {
        const int i = b * CDIM + c;
        const float v = (mask[i] != 0) ? sc[i] : NEG_INF;
        mx = fmaxf(mx, v);
    }
    red[tid] = mx;
    __syncthreads();
    for (int s = 128; s > 0; s >>= 1) {
        if (tid < s) red[tid] = fmaxf(red[tid], red[tid + s]);
        __syncthreads();
    }
    mx = red[0];
    __syncthreads();

    float sum = 0.f;
    for (int c = tid; c < CDIM; c += 256) {
        const int i = b * CDIM + c;
        const float e = (mask[i] != 0) ? __expf(sc[i] - mx) : 0.f;
        sc[i] = e;
        sum += e;
    }
    red[tid] = sum;
    __syncthreads();
    for (int s = 128; s > 0; s >>= 1) {
        if (tid < s) red[tid] += red[tid + s];
        __syncthreads();
    }
    const float inv = 1.f / red[0];
    for (int c = tid; c < CDIM; c += 256) sc[b * CDIM + c] *= inv;
}

// ---------------------------------------------------------------------------
// Kernel 3: out[b,g,c] = p[b,c] * sigmoid(g_h[b,g] + g_c[b,c] + bias)
// float4 streaming; dominates runtime (67MB write), pure bandwidth.
// ---------------------------------------------------------------------------
__global__ void fuse_out_kernel(const float* __restrict__ p,
                                const float* __restrict__ gc,
                                const float* __restrict__ gh,
                                const float* __restrict__ bgate,
                                float* __restrict__ out) {
    const int C4  = CDIM / 4;
    const int idx = blockIdx.x * blockDim.x + threadIdx.x;   // < B*G*C4
    const int c4  = idx % C4;
    const int g   = (idx / C4) % GDIM;
    const int b   = idx / (C4 * GDIM);

    const float h = gh[b * GDIM + g] + bgate[0];
    const float4 p4  = ((const float4*)p )[b * C4 + c4];
    const float4 gc4 = ((const float4*)gc)[b * C4 + c4];

    float4 o;
    o.x = p4.x * (1.f / (1.f + __expf(-(h + gc4.x))));
    o.y = p4.y * (1.f / (1.f + __expf(-(h + gc4.y))));
    o.z = p4.z * (1.f / (1.f + __expf(-(h + gc4.z))));
    o.w = p4.w * (1.f / (1.f + __expf(-(h + gc4.w))));
    ((float4*)out)[idx] = o;
}

// ---------------------------------------------------------------------------
extern "C" void kernel_launch(void* const* d_in, const int* in_sizes, int n_in,
                              void* d_out, int out_size, void* d_ws, size_t ws_size,
                              hipStream_t stream) {
    const float* hidden  = (const float*)d_in[0];   // (8,512,1024)
    const float* context = (const float*)d_in[1];   // (8,4096,1024)
    // d_in[2] = encoder_output: unused by the reference computation
    const float* w_attn  = (const float*)d_in[3];   // (2048,)
    const float* w_gate  = (const float*)d_in[4];   // (2048,)
    const float* b_gate  = (const float*)d_in[5];   // (1,)
    const int*   mask    = (const int*)d_in[6];     // (8,4096)
    float* out = (float*)d_out;                     // (8,512,4096)

    // Workspace layout (floats)
    float* ws  = (float*)d_ws;
    float* s_c = ws;                  // 8*4096  -> becomes softmax probs in place
    float* g_c = ws + 32768;          // 8*4096
    float* s_h = ws + 65536;          // 8*512   (dead: softmax cancels score_h)
    float* g_h = ws + 69632;          // 8*512

    const int Mc = BATCH * CDIM;      // 32768 context rows
    const int Mh = BATCH * GDIM;      // 4096 hidden rows

    // 8 waves/block, 16 rows/wave -> 128 rows/block; grids are exact.
    gemv2_wmma_kernel<<<Mc / 128, 256, 0, stream>>>(
        context, w_attn + HID, w_gate + HID, s_c, g_c);
    gemv2_wmma_kernel<<<Mh / 128, 256, 0, stream>>>(
        hidden, w_attn, w_gate, s_h, g_h);

    softmax_mask_kernel<<<BATCH, 256, 0, stream>>>(s_c, mask);

    const int total4 = BATCH * GDIM * (CDIM / 4);   // 4,194,304
    fuse_out_kernel<<<total4 / 256, 256, 0, stream>>>(s_c, g_c, g_h, b_gate, out);

    (void)in_sizes; (void)n_in; (void)out_size; (void)ws_size;
}